// Agent_9148280340767
// MI455X (gfx1250) — compile-verified
//
#include <hip/hip_runtime.h>
#include <math.h>

// Problem sizes (fixed by the reference)
#define Bsz 128
#define Nn  1024
#define Ee  256
#define Hh  16
#define Kk  16

typedef __attribute__((ext_vector_type(2))) float v2f;
typedef __attribute__((ext_vector_type(8))) float v8f;

// V_WMMA_F32_16X16X4_F32 : D(16x16,f32) += A(16x4,f32) x B(4x16,f32)
// Lanes 0-15 hold hwK slots {0,1} (VGPR0/1), lanes 16-31 hold {2,3}.
// Logical columns are remapped so each lane half owns a contiguous 8-column
// slice (base 0 / 8); A and B use the same mapping so the dot is exact.
// A rows are replicated (single query row) -> D rows identical -> every lane
// holds the result for column (lane&15) in acc[0]; duplicate stores are safe.
__device__ __forceinline__ v8f wmma4(v2f a, v2f b, v8f c) {
  return __builtin_amdgcn_wmma_f32_16x16x4_f32(false, a, false, b, (short)0, c,
                                               false, false);
}

// ---- gfx1250 async global->LDS copy (ASYNCcnt-tracked) --------------------
__device__ __forceinline__ uint32_t lds_off(const void* p) {
  return (uint32_t)(uintptr_t)p;  // LDS aperture: low 32 bits = LDS byte offset
}
__device__ __forceinline__ void async_ld_b128(uint32_t lds, uint64_t gaddr) {
  asm volatile("global_load_async_to_lds_b128 %0, %1, off"
               :: "v"(lds), "v"(gaddr) : "memory");
}
__device__ __forceinline__ void wait_async_le2() {
  asm volatile("s_wait_asynccnt 0x2" ::: "memory");
}
__device__ __forceinline__ void wait_async_0() {
  asm volatile("s_wait_asynccnt 0x0" ::: "memory");
}

// ---------------------------------------------------------------------------
// K1: query[b,e] = fixed_context[b,e] + cat(prev_emb, gdf)[b,:] @ W_cs[:,e]
// ---------------------------------------------------------------------------
__global__ void __launch_bounds__(256) k_query(
    const float* __restrict__ fixed_ctx, const float* __restrict__ prev_emb,
    const float* __restrict__ gdf, const float* __restrict__ Wcs,
    float* __restrict__ query) {
  int b = blockIdx.x, e = threadIdx.x;
  __shared__ float cs[Ee + 2];
  cs[e] = prev_emb[b * Ee + e];
  if (e < 2) cs[Ee + e] = gdf[b * 2 + e];
  __syncthreads();
  float acc = fixed_ctx[b * Ee + e];
#pragma unroll 4
  for (int i = 0; i < Ee + 2; ++i) acc = fmaf(cs[i], Wcs[i * Ee + e], acc);
  query[b * Ee + e] = acc;
}

// ---------------------------------------------------------------------------
// K2: per (b,h): compat -> masked softmax -> heads.  128 thr = 4 waves.
// Rank-3 trick: all dynamic-feature terms are applied as 3 precomputed
// scalars outside the streaming loops; inner loops are pure load + WMMA.
// ---------------------------------------------------------------------------
__global__ void __launch_bounds__(128) k_attn(
    const float* __restrict__ gK, const float* __restrict__ gV,
    const float* __restrict__ ndf, const float* __restrict__ Wn,
    const int* __restrict__ mask, const float* __restrict__ query,
    float* __restrict__ heads_out) {
  const int b = blockIdx.x >> 4, h = blockIdx.x & 15;
  const int tid = threadIdx.x, wave = tid >> 5, lane = tid & 31;
  const int nl = lane & 15;
  const int cbase = (lane >> 4) << 3;  // contiguous 8-slice base: 0 or 8
  const int rb = cbase;                // row base for heads tiles

  __shared__ __align__(16) float s_ndf[Nn * 3];      // 12 KB
  __shared__ __align__(16) float s_c[Nn];            // compat -> attn
  __shared__ float s_mb[Nn];                         // mask bias 0 / -3e38
  __shared__ __align__(16) float s_vbuf[4][2][256];  // V tiles, 2 slots/wave
  __shared__ float s_qh[Kk];
  __shared__ float s_wV[3][Kk];
  __shared__ float s_qw[3];
  __shared__ float s_hp[4][Kk];
  __shared__ float s_red[4];
  __shared__ float s_ra[4][3];

  // ---- stage ndf via async global->LDS (6 x 2KB) ----
  {
    uint32_t nb = lds_off(s_ndf) + tid * 16;
    uint64_t gb = (uint64_t)(uintptr_t)(ndf + (size_t)b * Nn * 3) + tid * 16;
#pragma unroll
    for (int it = 0; it < 6; ++it)
      async_ld_b128(nb + it * 2048, gb + it * 2048);
  }
  for (int i = tid; i < Nn; i += 128) {
    int mk = mask[b * Nn + i];
    s_mb[i] = mk ? 0.f : -3.0e38f;
  }
  if (tid < Kk) {
    s_qh[tid] = query[b * Ee + h * Kk + tid] * 0.25f;  // fold 1/sqrt(16)
    for (int f = 0; f < 3; ++f)
      s_wV[f][tid] = Wn[f * (3 * Ee) + h * Kk + tid];  // gV cols
  }
  wait_async_0();
  __syncthreads();

  // qw[f] = (q/4) . wK_f   (rank-3 compat correction)
  if (tid < 3) {
    const float* wkf = Wn + tid * (3 * Ee) + Ee + h * Kk;  // gK cols
    float t = 0.f;
#pragma unroll
    for (int k = 0; k < Kk; ++k) t = fmaf(s_qh[k], wkf[k], t);
    s_qw[tid] = t;
  }
  float qa[8];
#pragma unroll
  for (int j = 0; j < 8; ++j) qa[j] = s_qh[cbase + j];
  __syncthreads();
  const float qw0 = s_qw[0], qw1 = s_qw[1], qw2 = s_qw[2];

  // ---- compat: pure b128 loads + WMMA, 1-deep pipeline ----
  const float* Kst = gK + ((size_t)h * Bsz + b) * Nn * Kk;
  float4 ka, kb;
  {
    const int row = wave * 16 + nl;
    ka = *(const float4*)(Kst + (size_t)row * Kk + cbase);
    kb = *(const float4*)(Kst + (size_t)row * Kk + cbase + 4);
  }
  for (int nt = wave; nt < Nn / 16; nt += 4) {
    const int nx = (nt + 4 < Nn / 16) ? nt + 4 : nt;  // clamped prefetch
    const int rowx = nx * 16 + nl;
    float4 kc = *(const float4*)(Kst + (size_t)rowx * Kk + cbase);
    float4 kd = *(const float4*)(Kst + (size_t)rowx * Kk + cbase + 4);
    v8f acc = {};
#define CSTEP(ax, ay, bx, by)                                   \
    { v2f a; a.x = (ax); a.y = (ay);                            \
      v2f bb; bb.x = (bx); bb.y = (by);                         \
      acc = wmma4(a, bb, acc); }
    CSTEP(qa[0], qa[1], ka.x, ka.y)
    CSTEP(qa[2], qa[3], ka.z, ka.w)
    CSTEP(qa[4], qa[5], kb.x, kb.y)
    CSTEP(qa[6], qa[7], kb.z, kb.w)
#undef CSTEP
    const int rq = nt * 16 + nl;
    float cr = fmaf(s_ndf[rq * 3], qw0, acc[0]);
    cr = fmaf(s_ndf[rq * 3 + 1], qw1, cr);
    cr = fmaf(s_ndf[rq * 3 + 2], qw2, cr);
    s_c[rq] = cr;  // all 32 lanes: duplicate identical write, branch-free
    ka = kc; kb = kd;
  }
  __syncthreads();

  // ---- masked softmax + attn-weighted ndf sums (register-resident) ----
  float cv[8];
#pragma unroll
  for (int j = 0; j < 8; ++j) {
    const int i = tid + j * 128;
    cv[j] = s_c[i] + s_mb[i];
  }
  float m = cv[0];
#pragma unroll
  for (int j = 1; j < 8; ++j) m = fmaxf(m, cv[j]);
  for (int off = 16; off; off >>= 1) m = fmaxf(m, __shfl_xor(m, off, 32));
  if (lane == 0) s_red[wave] = m;
  __syncthreads();
  m = fmaxf(fmaxf(s_red[0], s_red[1]), fmaxf(s_red[2], s_red[3]));
  float ev[8], s = 0.f;
#pragma unroll
  for (int j = 0; j < 8; ++j) { ev[j] = __expf(cv[j] - m); s += ev[j]; }
  for (int off = 16; off; off >>= 1) s += __shfl_xor(s, off, 32);
  __syncthreads();  // everyone done reading s_red (max)
  if (lane == 0) s_red[wave] = s;
  __syncthreads();
  const float inv = 1.f / (s_red[0] + s_red[1] + s_red[2] + s_red[3]);
  float a0 = 0.f, a1 = 0.f, a2 = 0.f;
#pragma unroll
  for (int j = 0; j < 8; ++j) {
    const int i = tid + j * 128;
    const float at = ev[j] * inv;
    s_c[i] = at;
    a0 = fmaf(at, s_ndf[i * 3 + 0], a0);
    a1 = fmaf(at, s_ndf[i * 3 + 1], a1);
    a2 = fmaf(at, s_ndf[i * 3 + 2], a2);
  }
  for (int off = 16; off; off >>= 1) {
    a0 += __shfl_xor(a0, off, 32);
    a1 += __shfl_xor(a1, off, 32);
    a2 += __shfl_xor(a2, off, 32);
  }
  if (lane == 0) { s_ra[wave][0] = a0; s_ra[wave][1] = a1; s_ra[wave][2] = a2; }
  __syncthreads();

  // ---- heads: async double-buffered V tiles, pure LDS loads + WMMA ----
  const float* Vst = gV + ((size_t)h * Bsz + b) * Nn * Kk;
  const uint32_t vb[2] = {lds_off(&s_vbuf[wave][0][0]),
                          lds_off(&s_vbuf[wave][1][0])};
  {
    uint64_t g = (uint64_t)(uintptr_t)(Vst + (size_t)wave * 256) + lane * 16;
    async_ld_b128(vb[0] + lane * 16, g);
    async_ld_b128(vb[0] + lane * 16 + 512, g + 512);
  }
  v8f hacc = {};
  int slot = 0;
  for (int nt = wave; nt < Nn / 16; nt += 4) {
    const int nx = (nt + 4 < Nn / 16) ? nt + 4 : nt;
    {
      uint64_t g = (uint64_t)(uintptr_t)(Vst + (size_t)nx * 256) + lane * 16;
      async_ld_b128(vb[slot ^ 1] + lane * 16, g);
      async_ld_b128(vb[slot ^ 1] + lane * 16 + 512, g + 512);
    }
    wait_async_le2();  // oldest pair (current slot) complete
    const float* vt = &s_vbuf[wave][slot][0];
    const int tb = nt * 16;
    float4 at0 = *(const float4*)&s_c[tb + rb];
    float4 at1 = *(const float4*)&s_c[tb + rb + 4];
#define HSTEP(j, ax, ay)                                   \
    { v2f a; a.x = (ax); a.y = (ay);                       \
      v2f bb;                                              \
      bb.x = vt[(rb + 2 * j) * Kk + nl];                   \
      bb.y = vt[(rb + 2 * j + 1) * Kk + nl];               \
      hacc = wmma4(a, bb, hacc); }
    HSTEP(0, at0.x, at0.y)
    HSTEP(1, at0.z, at0.w)
    HSTEP(2, at1.x, at1.y)
    HSTEP(3, at1.z, at1.w)
#undef HSTEP
    slot ^= 1;
  }
  wait_async_0();
  s_hp[wave][nl] = hacc[0];  // duplicate identical write, branch-free
  __syncthreads();
  if (tid < Kk) {
    const float A0 = s_ra[0][0] + s_ra[1][0] + s_ra[2][0] + s_ra[3][0];
    const float A1 = s_ra[0][1] + s_ra[1][1] + s_ra[2][1] + s_ra[3][1];
    const float A2 = s_ra[0][2] + s_ra[1][2] + s_ra[2][2] + s_ra[3][2];
    float hv = s_hp[0][tid] + s_hp[1][tid] + s_hp[2][tid] + s_hp[3][tid];
    hv = fmaf(A0, s_wV[0][tid], hv);
    hv = fmaf(A1, s_wV[1][tid], hv);
    hv = fmaf(A2, s_wV[2][tid], hv);
    heads_out[b * Ee + h * Kk + tid] = hv;
  }
}

// ---------------------------------------------------------------------------
// K3: per (b, 128-node chunk): fq = heads @ W_out (cheap),
//     logits[n] = tanh( (fq.LK[n] + ndf[n].t) / 16 ) * 10   via WMMA,
//     with t_f = fq . Wl_f  (rank-3 dynamic-logit correction).
// ---------------------------------------------------------------------------
__global__ void __launch_bounds__(256) k_logits(
    const float* __restrict__ LK, const float* __restrict__ ndf,
    const float* __restrict__ Wn, const float* __restrict__ Wout,
    const float* __restrict__ heads, float* __restrict__ logits) {
  const int b = blockIdx.x >> 3, chunk = blockIdx.x & 7;
  const int tid = threadIdx.x, wave = tid >> 5, lane = tid & 31;
  const int nl = lane & 15;
  const int base = (lane >> 4) << 3;  // 0 or 8

  __shared__ float s_h[Ee];
  __shared__ __align__(16) float s_fq[Ee];
  __shared__ float s_nd[128 * 3];
  __shared__ float s_part[48];
  __shared__ float s_tl[3];

  s_h[tid] = heads[b * Ee + tid];
  for (int i = tid; i < 128 * 3; i += 256)
    s_nd[i] = ndf[(size_t)b * Nn * 3 + chunk * 384 + i];
  __syncthreads();

  float acc = 0.f;
#pragma unroll 4
  for (int i = 0; i < Ee; ++i) acc = fmaf(s_h[i], Wout[i * Ee + tid], acc);
  s_fq[tid] = acc;
  __syncthreads();

  // t_f = fq . Wl_f : 48 partial sums of 16, then 3 final sums
  if (tid < 48) {
    const int f = tid >> 4, seg = (tid & 15) * 16;
    const float* w = Wn + f * (3 * Ee) + 2 * Ee + seg;
    float t = 0.f;
#pragma unroll
    for (int e = 0; e < 16; ++e) t = fmaf(s_fq[seg + e], w[e], t);
    s_part[tid] = t;
  }
  __syncthreads();
  if (tid < 3) {
    float t = 0.f;
#pragma unroll
    for (int j = 0; j < 16; ++j) t += s_part[tid * 16 + j];
    s_tl[tid] = t;
  }
  __syncthreads();
  const float t0 = s_tl[0], t1 = s_tl[1], t2 = s_tl[2];

  const int n0 = chunk * 128 + wave * 16;
  const int row = wave * 16 + nl;
  const float f0 = s_nd[row * 3], f1 = s_nd[row * 3 + 1],
              f2 = s_nd[row * 3 + 2];
  const float* lkr = LK + ((size_t)b * Nn + n0 + nl) * Ee;

  float4 c0 = *(const float4*)(lkr + base);
  float4 c1 = *(const float4*)(lkr + base + 4);
  v8f d = {};
  for (int g = 0; g < 16; ++g) {
    const int gn = (g < 15) ? g + 1 : g;  // clamped prefetch
    float4 p0 = *(const float4*)(lkr + gn * 16 + base);
    float4 p1 = *(const float4*)(lkr + gn * 16 + base + 4);
    const int cb = g * 16 + base;
    float4 qa0 = *(const float4*)&s_fq[cb];
    float4 qa1 = *(const float4*)&s_fq[cb + 4];
#define LSTEP(ax, ay, bx, by)                                   \
    { v2f a; a.x = (ax); a.y = (ay);                            \
      v2f bb; bb.x = (bx); bb.y = (by);                         \
      d = wmma4(a, bb, d); }
    LSTEP(qa0.x, qa0.y, c0.x, c0.y)
    LSTEP(qa0.z, qa0.w, c0.z, c0.w)
    LSTEP(qa1.x, qa1.y, c1.x, c1.y)
    LSTEP(qa1.z, qa1.w, c1.z, c1.w)
#undef LSTEP
    c0 = p0; c1 = p1;
  }
  float lg = d[0];
  lg = fmaf(f0, t0, lg); lg = fmaf(f1, t1, lg); lg = fmaf(f2, t2, lg);
  lg = tanhf(lg * (1.0f / 16.0f)) * 10.0f;
  logits[(size_t)b * Nn + n0 + nl] = lg;  // duplicate identical write
}

// ---------------------------------------------------------------------------
// K4: per b: masked max/argmax (first occurrence) + log-sum-exp. Branch-free.
// ---------------------------------------------------------------------------
__global__ void __launch_bounds__(256) k_select(
    const float* __restrict__ logits, const int* __restrict__ mask,
    float* __restrict__ out, float* __restrict__ prob) {
  const int b = blockIdx.x, tid = threadIdx.x;
  __shared__ float sm[256];
  __shared__ int si[256];
  __shared__ float ss[256];
  const float* lr = logits + (size_t)b * Nn;
  const int* mr = mask + b * Nn;
  float lv[4], bias[4];
#pragma unroll
  for (int j = 0; j < 4; ++j) {
    const int i = tid + j * 256;
    lv[j] = lr[i];
    bias[j] = mr[i] ? 0.f : -3.0e38f;
  }
  float m = -3.0e38f;
  int mi = Nn - 1;
#pragma unroll
  for (int j = 0; j < 4; ++j) {
    const float v = lv[j] + bias[j];
    if (v > m) { m = v; mi = tid + j * 256; }
  }
  sm[tid] = m; si[tid] = mi;
  __syncthreads();
  for (int off = 128; off; off >>= 1) {
    if (tid < off) {
      float om = sm[tid + off]; int oi = si[tid + off];
      if (om > sm[tid] || (om == sm[tid] && oi < si[tid])) {
        sm[tid] = om; si[tid] = oi;
      }
    }
    __syncthreads();
  }
  m = sm[0]; mi = si[0];
  float s = 0.f;
#pragma unroll
  for (int j = 0; j < 4; ++j) s += __expf(lv[j] + bias[j] - m);
  ss[tid] = s;
  __syncthreads();
  for (int off = 128; off; off >>= 1) {
    if (tid < off) ss[tid] += ss[tid + off];
    __syncthreads();
  }
  if (tid == 0) {
    const float lp = -__logf(ss[0]);  // = l_max - logsumexp
    out[b] = (float)mi;
    out[Bsz + b] = lp;
    prob[b] = __expf(lp);
  }
}

// K5: entropy = -sum_b prob[b]*logprob[b]  (deterministic tree reduce)
__global__ void __launch_bounds__(128) k_entropy(
    const float* __restrict__ lp, const float* __restrict__ prob,
    float* __restrict__ out) {
  const int tid = threadIdx.x;
  __shared__ float r[128];
  r[tid] = -prob[tid] * lp[tid];
  __syncthreads();
  for (int off = 64; off; off >>= 1) {
    if (tid < off) r[tid] += r[tid + off];
    __syncthreads();
  }
  if (tid == 0) out[2 * Bsz] = r[0];
}

extern "C" void kernel_launch(void* const* d_in, const int* in_sizes, int n_in,
                              void* d_out, int out_size, void* d_ws,
                              size_t ws_size, hipStream_t stream) {
  (void)in_sizes; (void)n_in; (void)out_size; (void)ws_size;
  // d_in[0] = item_embeddings : unused by the reference math (shape only).
  const float* fixed_ctx = (const float*)d_in[1];
  const float* prev_emb  = (const float*)d_in[2];
  const float* ndf       = (const float*)d_in[3];
  const float* gdf       = (const float*)d_in[4];
  const float* gV        = (const float*)d_in[5];
  const float* gK        = (const float*)d_in[6];
  const float* LK        = (const float*)d_in[7];
  const int*   mask      = (const int*)d_in[8];
  const float* Wcs       = (const float*)d_in[9];
  const float* Wn        = (const float*)d_in[10];
  const float* Wout      = (const float*)d_in[11];
  float* out = (float*)d_out;

  float* ws     = (float*)d_ws;
  float* query  = ws;                            // B*E
  float* heads  = ws + Bsz * Ee;                 // B*E
  float* logits = ws + 2 * Bsz * Ee;             // B*N
  float* prob   = ws + 2 * Bsz * Ee + Bsz * Nn;  // B

  k_query<<<dim3(Bsz), dim3(Ee), 0, stream>>>(fixed_ctx, prev_emb, gdf, Wcs,
                                              query);
  k_attn<<<dim3(Bsz * Hh), dim3(128), 0, stream>>>(gK, gV, ndf, Wn, mask,
                                                   query, heads);
  k_logits<<<dim3(Bsz * 8), dim3(256), 0, stream>>>(LK, ndf, Wn, Wout, heads,
                                                    logits);
  k_select<<<dim3(Bsz), dim3(256), 0, stream>>>(logits, mask, out, prob);
  k_entropy<<<dim3(1), dim3(Bsz), 0, stream>>>(out + Bsz, prob, out);
}